// GlobalSelfAttention_9337258902217
// MI455X (gfx1250) — compile-verified
//
#include <hip/hip_runtime.h>

typedef float v2f __attribute__((ext_vector_type(2)));
typedef float v4f __attribute__((ext_vector_type(4)));
typedef float v8f __attribute__((ext_vector_type(8)));
typedef int   v4i __attribute__((ext_vector_type(4)));

// Problem dims
constexpr int BATCH = 32;
constexpr int SEQ   = 256;
constexpr int EMB   = 32;
constexpr int DH    = 32;
constexpr int FIN   = SEQ * EMB;   // 8192
constexpr int FOUT  = SEQ * DH;    // 8192

#define WMMA4(a, bfr, c) \
    __builtin_amdgcn_wmma_f32_16x16x4_f32(false, (a), false, (bfr), (short)0, (c), false, false)

// --- CDNA5 async global->LDS copy path (guarded: falls back to sync staging) ---
#if __has_builtin(__builtin_amdgcn_global_load_async_to_lds_b128)
#define HAS_ASYNC_LDS 1
typedef __attribute__((address_space(1))) v4i* gv4i_p;   // global int4*
typedef __attribute__((address_space(3))) v4i* lv4i_p;   // LDS int4*
#define GASYNC_B128(gsrc, ldst)                                                \
    __builtin_amdgcn_global_load_async_to_lds_b128(                            \
        (gv4i_p)(v4i*)(void*)const_cast<float*>(gsrc),                         \
        (lv4i_p)(v4i*)(void*)(ldst), 0, 0)
#else
#define HAS_ASYNC_LDS 0
#endif

__device__ __forceinline__ void wait_async_lds()
{
#if HAS_ASYNC_LDS
#if __has_builtin(__builtin_amdgcn_s_wait_asynccnt)
    __builtin_amdgcn_s_wait_asynccnt(0);
#else
    asm volatile("s_wait_asynccnt 0x0" ::: "memory");
#endif
#endif
}

// ---------------------------------------------------------------------------
// Kernel 1: QKV projection.  out[m,n] = sum_k x[m,k]*W[n,k] + bias[n]
// M=32, N=K=8192, fp32 — HBM-bound on the 805 MB weight stream.
// grid = (FOUT/128, 3), block = 256 (8 waves). Wave w owns 16 output cols.
// Double-buffered LDS chunks; weights streamed via global_load_async_to_lds.
// ---------------------------------------------------------------------------
constexpr int KC        = 32;      // K-chunk per stage
constexpr int WT_STRIDE = KC + 4;  // 36 dwords: 16B-aligned rows, conflict-free frags
constexpr int XT_STRIDE = KC + 4;

__device__ __forceinline__ void stage_chunk(const float* __restrict__ W,
                                            const float* __restrict__ x,
                                            float* __restrict__ Wbuf,
                                            float* __restrict__ Xbuf,
                                            int nblk, int kc, int tid)
{
    // W tile: 128 rows x 32 k (1024 float4), X tile: 32 x 32 (256 float4)
    #pragma unroll
    for (int i = 0; i < 4; ++i) {
        int f4  = tid + i * 256;            // 0..1023
        int row = f4 >> 3;                  // 0..127
        int c4  = (f4 & 7) << 2;            // 0..28
        const float* g = W + (size_t)(nblk + row) * FIN + kc + c4;
        float* l = &Wbuf[row * WT_STRIDE + c4];
#if HAS_ASYNC_LDS
        GASYNC_B128(g, l);
#else
        *(v4f*)l = *(const v4f*)g;
#endif
    }
    {
        int row = tid >> 3;                 // 0..31
        int c4  = (tid & 7) << 2;
        const float* g = x + (size_t)row * FIN + kc + c4;
        float* l = &Xbuf[row * XT_STRIDE + c4];
#if HAS_ASYNC_LDS
        GASYNC_B128(g, l);
#else
        *(v4f*)l = *(const v4f*)g;
#endif
    }
}

__global__ __launch_bounds__(256)
void qkv_gemm_kernel(const float* __restrict__ x,
                     const float* __restrict__ Wq, const float* __restrict__ bq,
                     const float* __restrict__ Wk, const float* __restrict__ bk,
                     const float* __restrict__ Wv, const float* __restrict__ bv,
                     float* __restrict__ ws)
{
    __shared__ float Wt[2][128 * WT_STRIDE];  // 2 x 18.4 KB
    __shared__ float Xt[2][32  * XT_STRIDE];  // 2 x  4.6 KB

    const int tid   = threadIdx.x;
    const int wave  = tid >> 5;
    const int lane  = tid & 31;
    const int l16   = lane & 15;
    const int khalf = (lane >> 4) * 2;       // K sub-offset 0 or 2 (A/B frag layout)
    const int mhi   = (lane >> 4) << 3;      // C-layout row offset 0 or 8

    const int mat = blockIdx.y;
    const float* W    = (mat == 0) ? Wq : (mat == 1) ? Wk : Wv;
    const float* bias = (mat == 0) ? bq : (mat == 1) ? bk : bv;
    float* out = ws + (size_t)mat * (BATCH * FOUT);

    const int nblk = blockIdx.x * 128;

    v8f acc0 = {};  // C tile rows 0..15
    v8f acc1 = {};  // C tile rows 16..31

    // prologue: stage chunk 0 into buffer 0
    stage_chunk(W, x, &Wt[0][0], &Xt[0][0], nblk, 0, tid);

    int p = 0;
    for (int kc = 0; kc < FIN; kc += KC, p ^= 1) {
        wait_async_lds();       // my buf-p copies are complete
        __syncthreads();        // everyone's buf-p complete; buf-p^1 reads (prev iter) done

        if (kc + KC < FIN)      // overlap: stage next chunk while computing this one
            stage_chunk(W, x, &Wt[p ^ 1][0], &Xt[p ^ 1][0], nblk, kc + KC, tid);

        const float* Wp = &Wt[p][0];
        const float* Xp = &Xt[p][0];
        #pragma unroll
        for (int j = 0; j < KC / 4; ++j) {
            int k = j * 4 + khalf;
            v2f a0 = *(const v2f*)(&Xp[l16 * XT_STRIDE + k]);
            v2f a1 = *(const v2f*)(&Xp[(16 + l16) * XT_STRIDE + k]);
            v2f bf = *(const v2f*)(&Wp[(wave * 16 + l16) * WT_STRIDE + k]);
            acc0 = WMMA4(a0, bf, acc0);
            acc1 = WMMA4(a1, bf, acc1);
        }
    }

    const int n  = nblk + wave * 16 + l16;
    const float bn = bias[n];
    #pragma unroll
    for (int v = 0; v < 8; ++v) {
        int m0 = v + mhi;   // rows v / v+8 per C layout
        out[(size_t)m0 * FOUT + n]        = acc0[v] + bn;
        out[(size_t)(m0 + 16) * FOUT + n] = acc1[v] + bn;
    }
}

// ---------------------------------------------------------------------------
// Kernel 2: per-batch attention with unstabilized softmax.
//   Z[m,d] = coeff * (sum_t exp(s_mt) V[t,d]) / (sum_t exp(s_mt) + 1e-5)
// Single sweep over t-tiles; denominator via WMMA against all-ones B operand
// so num and den land in identical C layouts (lane-local finalize).
// grid = BATCH, block = 256 (8 waves, 32 rows each).
// ---------------------------------------------------------------------------
constexpr int VT_STRIDE = SEQ + 2;  // 258
constexpr int P_STRIDE  = 18;

__global__ __launch_bounds__(256)
void attention_kernel(const float* __restrict__ ws, float* __restrict__ out)
{
    __shared__ float Vt[DH * VT_STRIDE];     // V transposed: 33 KB
    __shared__ float Pw[8][16 * P_STRIDE];   // per-wave P-tile scratch: 9.2 KB

    const int b     = blockIdx.x;
    const int tid   = threadIdx.x;
    const int wave  = tid >> 5;
    const int lane  = tid & 31;
    const int l16   = lane & 15;
    const int khalf = (lane >> 4) * 2;
    const int mhi   = (lane >> 4) << 3;

    const float* Q  = ws + (size_t)b * (SEQ * DH);
    const float* Km = ws + (size_t)(BATCH * FOUT) + (size_t)b * (SEQ * DH);
    const float* V  = ws + (size_t)(2 * BATCH * FOUT) + (size_t)b * (SEQ * DH);

    // stage V transposed into LDS (coalesced global read, scatter to LDS)
    #pragma unroll
    for (int i = 0; i < 8; ++i) {
        int f4  = tid + i * 256;           // 0..2047 float4s
        int row = f4 >> 3;                 // t index 0..255
        int col = (f4 & 7) << 2;           // d index 0..28
        v4f v = *(const v4f*)(V + (size_t)f4 * 4);
        Vt[(col + 0) * VT_STRIDE + row] = v.x;
        Vt[(col + 1) * VT_STRIDE + row] = v.y;
        Vt[(col + 2) * VT_STRIDE + row] = v.z;
        Vt[(col + 3) * VT_STRIDE + row] = v.w;
    }
    __syncthreads();

    const float coeff = 0.17677669529663689f;  // 1/sqrt(32)
    const v2f one2 = {1.0f, 1.0f};
    float* Pme = &Pw[wave][0];

    for (int mt = wave * 2; mt < wave * 2 + 2; ++mt) {
        // preload Q A-fragments for this 16-row tile (constant across t sweep)
        v2f qf[8];
        #pragma unroll
        for (int j = 0; j < 8; ++j)
            qf[j] = *(const v2f*)(Q + (size_t)(mt * 16 + l16) * DH + j * 4 + khalf);

        v8f num0 = {}, num1 = {}, den = {};

        for (int tt = 0; tt < SEQ / 16; ++tt) {
            // S tile = Q(16xD) * K(16xD)^T  via 8 x wmma K=4
            v8f s = {};
            #pragma unroll
            for (int j = 0; j < 8; ++j) {
                v2f kf = *(const v2f*)(Km + (size_t)(tt * 16 + l16) * DH + j * 4 + khalf);
                s = WMMA4(qf[j], kf, s);
            }
            // P = exp(S)  (reference uses no max subtraction)
            v8f p;
            #pragma unroll
            for (int v = 0; v < 8; ++v) p[v] = __expf(s[v]);

            // C layout -> row-major LDS (A-layout source for next GEMM)
            #pragma unroll
            for (int v = 0; v < 8; ++v)
                Pme[(v + mhi) * P_STRIDE + l16] = p[v];
            asm volatile("s_wait_dscnt 0x0" ::: "memory");

            // num += P*V (two d-tiles), den += P*ones : K=16 -> 4 k-steps
            #pragma unroll
            for (int j2 = 0; j2 < 4; ++j2) {
                int k = j2 * 4 + khalf;
                v2f pa  = *(const v2f*)(&Pme[l16 * P_STRIDE + k]);
                v2f vb0 = *(const v2f*)(&Vt[l16 * VT_STRIDE + tt * 16 + k]);
                v2f vb1 = *(const v2f*)(&Vt[(16 + l16) * VT_STRIDE + tt * 16 + k]);
                num0 = WMMA4(pa, vb0, num0);
                num1 = WMMA4(pa, vb1, num1);
                den  = WMMA4(pa, one2, den);
            }
        }

        // Z = coeff * num / (den + 1e-5): den row-sum replicated across cols,
        // same C layout as num -> lane-local finalize.
        #pragma unroll
        for (int v = 0; v < 8; ++v) {
            int m = mt * 16 + v + mhi;
            float inv = coeff / (den[v] + 1e-5f);
            out[(size_t)b * (SEQ * DH) + (size_t)m * DH + l16]      = num0[v] * inv;
            out[(size_t)b * (SEQ * DH) + (size_t)m * DH + 16 + l16] = num1[v] * inv;
        }
    }
}

// ---------------------------------------------------------------------------
extern "C" void kernel_launch(void* const* d_in, const int* in_sizes, int n_in,
                              void* d_out, int out_size, void* d_ws, size_t ws_size,
                              hipStream_t stream)
{
    const float* x  = (const float*)d_in[0];
    const float* Wq = (const float*)d_in[1];
    const float* bq = (const float*)d_in[2];
    const float* Wk = (const float*)d_in[3];
    const float* bk = (const float*)d_in[4];
    const float* Wv = (const float*)d_in[5];
    const float* bv = (const float*)d_in[6];
    float* ws  = (float*)d_ws;   // needs 3 * 32 * 8192 * 4B = 3 MB for Q,K,V
    float* out = (float*)d_out;

    dim3 g1(FOUT / 128, 3);
    qkv_gemm_kernel<<<g1, 256, 0, stream>>>(x, Wq, bq, Wk, bk, Wv, bv, ws);
    attention_kernel<<<dim3(BATCH), 256, 0, stream>>>(ws, out);
}